// DownNet_75393855914259
// MI455X (gfx1250) — compile-verified
//
#include <hip/hip_runtime.h>
#include <hip/hip_bf16.h>

// ---------------------------------------------------------------------------
// Dense GCN + TopK pooling pipeline for MI455X (gfx1250).
// All heavy matmuls go through V_WMMA_F32_16X16X4_F32 (f32 WMMA, wave32).
// ---------------------------------------------------------------------------

typedef float v2f __attribute__((ext_vector_type(2)));
typedef float v4f __attribute__((ext_vector_type(4)));
typedef float v8f __attribute__((ext_vector_type(8)));

__device__ __forceinline__ v8f wmma_f32(v2f a, v2f b, v8f c) {
  // (neg_a, A, neg_b, B, c_mod, C, reuse_a, reuse_b)
  return __builtin_amdgcn_wmma_f32_16x16x4_f32(false, a, false, b, (short)0, c,
                                               false, false);
}

#define BM 64
#define BN 128
#define BK 16

// LDS pair-packed layout: row r holds K-pair (2r, 2r+1) interleaved per m/n so
// a WMMA fragment is one aligned 8-byte LDS load. Row stride padded so the two
// lane-halves (adjacent k-pair rows) hit disjoint bank sets.
#define ASTR2 ((BM + 16) * 2)   // floats per A pair-row
#define BSTR2 ((BN + 16) * 2)   // floats per B pair-row

// C[M,N] = A[M,K] @ B[K,N]; row-major, runtime leading dims.
// Tile fullness is uniform per block -> scalar branch between an unguarded
// b128 fast path and a guarded ragged path (edge blocks / last k-tile only).
__global__ __launch_bounds__(256) void k_gemm(const float* __restrict__ A,
                                              const float* __restrict__ B,
                                              float* __restrict__ C, int M,
                                              int N, int K, int lda, int ldb,
                                              int ldc) {
  __shared__ float As[(BK / 2) * ASTR2];   // 5.0 KB
  __shared__ float Bs[(BK / 2) * BSTR2];   // 9.0 KB

  const int t = threadIdx.x;
  const int lane = t & 31;
  const int wave = t >> 5;           // 8 waves
  const int wrow = wave >> 2;        // 0..1 -> 32-row strip
  const int wcol = wave & 3;         // 0..3 -> 32-col strip
  const int half = lane >> 4;        // lane half selects K pair / M+8
  const int l16 = lane & 15;

  const int m0 = blockIdx.y * BM;
  const int n0 = blockIdx.x * BN;

  // global-fill coordinates
  const int tm = t >> 2;             // 0..63   (A tile row)
  const int tk4 = (t & 3) << 2;      // 0,4,8,12 (A tile k base)
  const int bk = t >> 5;             // 0..7    (B tile k rows, +8 second pass)
  const int bn4 = (t & 31) << 2;     // 0..124  (B tile col base)

  const bool fullM = (m0 + BM <= M);       // uniform
  const bool fullN = (n0 + BN <= N);       // uniform

  v8f acc[2][2]{};

  for (int kt = 0; kt < K; kt += BK) {
    const bool fullK = (kt + BK <= K);     // uniform

    // ---- A tile 64x16: one float4 per thread, pair-packed into LDS ----
    {
      const float* ap = &A[(size_t)(m0 + tm) * lda + (kt + tk4)];
      v4f av;
      if (fullM && fullK) {                // uniform fast path
        av = *(const v4f*)ap;
        if (kt + 2 * BK <= K) __builtin_prefetch(ap + BK, 0, 1);
      } else {
        int gr = m0 + tm, gk = kt + tk4;
        av.x = (gr < M && gk + 0 < K) ? ap[0] : 0.0f;
        av.y = (gr < M && gk + 1 < K) ? ap[1] : 0.0f;
        av.z = (gr < M && gk + 2 < K) ? ap[2] : 0.0f;
        av.w = (gr < M && gk + 3 < K) ? ap[3] : 0.0f;
      }
      float* dst = &As[(tk4 >> 1) * ASTR2 + tm * 2];
      dst[0] = av.x;
      dst[1] = av.y;
      dst[ASTR2 + 0] = av.z;
      dst[ASTR2 + 1] = av.w;
    }
    // ---- B tile 16x128: two float4 per thread, pair-packed into LDS ----
#pragma unroll
    for (int i = 0; i < 2; ++i) {
      const int k = bk + 8 * i;            // 0..15
      const float* bp = &B[(size_t)(kt + k) * ldb + (n0 + bn4)];
      v4f bv;
      if (fullN && fullK) {                // uniform fast path
        bv = *(const v4f*)bp;
        if (kt + 2 * BK <= K) __builtin_prefetch(bp + (size_t)BK * ldb, 0, 1);
      } else {
        int gk = kt + k, gc = n0 + bn4;
        bv.x = (gk < K && gc + 0 < N) ? bp[0] : 0.0f;
        bv.y = (gk < K && gc + 1 < N) ? bp[1] : 0.0f;
        bv.z = (gk < K && gc + 2 < N) ? bp[2] : 0.0f;
        bv.w = (gk < K && gc + 3 < N) ? bp[3] : 0.0f;
      }
      float* dst = &Bs[(k >> 1) * BSTR2 + bn4 * 2 + (k & 1)];
      dst[0] = bv.x;
      dst[2] = bv.y;
      dst[4] = bv.z;
      dst[6] = bv.w;
    }
    __syncthreads();

#pragma unroll
    for (int kk = 0; kk < BK; kk += 4) {
      // A frag 16x4: lanes 0-15 K=kk+0,kk+1; lanes 16-31 K=kk+2,kk+3
      // => pair-row (kk>>1)+half, one b64 load per fragment.
      const int k2 = (kk >> 1) + half;
      const float* ap = &As[k2 * ASTR2 + (wrow * 32 + l16) * 2];
      v2f a0 = *(const v2f*)ap;
      v2f a1 = *(const v2f*)(ap + 32);
      const float* bp = &Bs[k2 * BSTR2 + (wcol * 32 + l16) * 2];
      v2f b0 = *(const v2f*)bp;
      v2f b1 = *(const v2f*)(bp + 32);
      acc[0][0] = wmma_f32(a0, b0, acc[0][0]);
      acc[0][1] = wmma_f32(a0, b1, acc[0][1]);
      acc[1][0] = wmma_f32(a1, b0, acc[1][0]);
      acc[1][1] = wmma_f32(a1, b1, acc[1][1]);
    }
    __syncthreads();
  }

  // C/D layout: VGPR v -> row v + 8*half, col = lane&15
#pragma unroll
  for (int i = 0; i < 2; ++i)
#pragma unroll
    for (int j = 0; j < 2; ++j)
#pragma unroll
      for (int v = 0; v < 8; ++v) {
        int row = m0 + wrow * 32 + i * 16 + v + 8 * half;
        int col = n0 + wcol * 32 + j * 16 + l16;
        if (row < M && col < N) C[(size_t)row * ldc + col] = acc[i][j][v];
      }
}

// A[dst, src] += 1 for each edge (coalesced-COO duplicate summation)
__global__ void k_scatter(const int* __restrict__ ei, int E,
                          float* __restrict__ A, int lda) {
  int e = blockIdx.x * blockDim.x + threadIdx.x;
  if (e < E) {
    int src = ei[e];
    int dst = ei[E + e];
    atomicAdd(&A[(size_t)dst * lda + src], 1.0f);
  }
}

// C = C + 2A off-diagonal, 0 on diagonal  (augment epilogue: (A+I)^2 - diag)
__global__ void k_augfix(float* __restrict__ C, const float* __restrict__ A,
                         int n, int lda) {
  int c = blockIdx.x * blockDim.x + threadIdx.x;
  int r = blockIdx.y;
  if (c < n) {
    size_t o = (size_t)r * lda + c;
    C[o] = (r == c) ? 0.0f : C[o] + 2.0f * A[o];
  }
}

// dinv[i] = (rowsum(A)_i + 2)^-1/2   (improved GCN self-loop weight 2)
__global__ __launch_bounds__(256) void k_dinv(const float* __restrict__ A,
                                              int n, int lda,
                                              float* __restrict__ dinv) {
  __shared__ float red[256];
  int row = blockIdx.x;
  float s = 0.0f;
  for (int j = threadIdx.x; j < n; j += 256) s += A[(size_t)row * lda + j];
  red[threadIdx.x] = s;
  __syncthreads();
  for (int off = 128; off > 0; off >>= 1) {
    if (threadIdx.x < off) red[threadIdx.x] += red[threadIdx.x + off];
    __syncthreads();
  }
  if (threadIdx.x == 0) dinv[row] = 1.0f / sqrtf(red[0] + 2.0f);
}

// Y[i,:] = dinv[i] * XW[i,:]
__global__ void k_rowscale(const float* __restrict__ XW,
                           const float* __restrict__ dinv,
                           float* __restrict__ Y, int c) {
  int col = blockIdx.x * blockDim.x + threadIdx.x;
  int row = blockIdx.y;
  if (col < c) {
    size_t o = (size_t)row * c + col;
    Y[o] = XW[o] * dinv[row];
  }
}

// X[i,c] = act( dinv[i]*(Z[i,c] + 2*Y[i,c]) + b[c] )
__global__ void k_gcnout(const float* __restrict__ Z,
                         const float* __restrict__ Y,
                         const float* __restrict__ dinv,
                         const float* __restrict__ b, float* __restrict__ X,
                         int c, int ldx, int do_relu) {
  int col = blockIdx.x * blockDim.x + threadIdx.x;
  int row = blockIdx.y;
  if (col < c) {
    size_t o = (size_t)row * c + col;
    float v = dinv[row] * (Z[o] + 2.0f * Y[o]) + b[col];
    if (do_relu) v = v > 0.0f ? v : 0.0f;
    X[(size_t)row * ldx + col] = v;
  }
}

// score[i] = tanh( (X[i,:] . p) / ||p|| )
__global__ __launch_bounds__(256) void k_score(const float* __restrict__ X,
                                               int n, int ldx,
                                               const float* __restrict__ p,
                                               int c,
                                               float* __restrict__ score) {
  __shared__ float ps[256];
  for (int i = threadIdx.x; i < c; i += 256) ps[i] = p[i];
  __syncthreads();
  float nrm = 0.0f;
  for (int j = 0; j < c; ++j) nrm += ps[j] * ps[j];
  nrm = sqrtf(nrm);
  int node = blockIdx.x * 256 + threadIdx.x;
  if (node < n) {
    float d = 0.0f;
    const float* xr = X + (size_t)node * ldx;
    for (int j = 0; j < c; ++j) d += xr[j] * ps[j];
    score[node] = tanhf(d / nrm);
  }
}

// Exact stable top-k: rank_i = #{j: s_j>s_i or (s_j==s_i and j<i)}; keep rank<k
__global__ __launch_bounds__(256) void k_select(const float* __restrict__ score,
                                                int n, int k,
                                                int* __restrict__ perm,
                                                float* __restrict__ vals) {
  __shared__ float ss[4096];
  for (int j = threadIdx.x; j < n; j += 256) ss[j] = score[j];
  __syncthreads();
  int i = blockIdx.x * 256 + threadIdx.x;
  if (i < n) {
    float si = ss[i];
    int rank = 0;
    for (int j = 0; j < n; ++j) {
      float sj = ss[j];
      rank += (sj > si) || (sj == si && j < i);
    }
    if (rank < k) {
      perm[rank] = i;
      vals[rank] = si;
    }
  }
}

// Xo[r,:] = X[perm[r],:] * vals[r]
__global__ void k_gatherx(const float* __restrict__ X, int ldx,
                          const int* __restrict__ perm,
                          const float* __restrict__ vals,
                          float* __restrict__ Xo, int ldo, int c) {
  int col = blockIdx.x * blockDim.x + threadIdx.x;
  int r = blockIdx.y;
  if (col < c)
    Xo[(size_t)r * ldo + col] = X[(size_t)perm[r] * ldx + col] * vals[r];
}

// Ao[r,c] = A[perm[r], perm[c]]
__global__ void k_gatherA(const float* __restrict__ A,
                          const int* __restrict__ perm, float* __restrict__ Ao,
                          int k, int lda) {
  int c = blockIdx.x * blockDim.x + threadIdx.x;
  int r = blockIdx.y;
  if (c < k)
    Ao[(size_t)r * lda + c] = A[(size_t)perm[r] * lda + perm[c]];
}

// ---------------------------------------------------------------------------

extern "C" void kernel_launch(void* const* d_in, const int* in_sizes, int n_in,
                              void* d_out, int out_size, void* d_ws,
                              size_t ws_size, hipStream_t stream) {
  (void)in_sizes; (void)n_in; (void)out_size; (void)ws_size;

  constexpr int NN = 4096, EE = 131072, CIN = 128, CH = 256, COUT = 128;
  constexpr int LDA = 4096;                 // fixed leading dim for all A bufs
  constexpr int K1 = 3277;                  // ceil(0.8*4096)
  constexpr int K2 = 1967;                  // ceil(0.6*3277)

  const float* x_in = (const float*)d_in[0];
  const int* ei = (const int*)d_in[1];
  const float* W0 = (const float*)d_in[2];
  const float* b0 = (const float*)d_in[3];
  const float* W1 = (const float*)d_in[4];
  const float* b1 = (const float*)d_in[5];
  const float* W2 = (const float*)d_in[6];
  const float* b2 = (const float*)d_in[7];
  const float* p1 = (const float*)d_in[8];
  const float* p2 = (const float*)d_in[9];
  const float* Wout = (const float*)d_in[10];
  const float* bout = (const float*)d_in[11];
  float* out = (float*)d_out;

  // workspace carve-up (~155.3 MB)
  char* ws = (char*)d_ws;
  const size_t szA = (size_t)NN * LDA * sizeof(float);   // 64 MB
  const size_t szX = (size_t)NN * CH * sizeof(float);    // 4 MB
  float* bufA = (float*)(ws);
  float* bufB = (float*)(ws + szA);
  float* X1   = (float*)(ws + 2 * szA);
  float* X2   = (float*)(ws + 2 * szA + szX);
  float* XW   = (float*)(ws + 2 * szA + 2 * szX);
  float* Y    = (float*)(ws + 2 * szA + 3 * szX);
  float* Z    = (float*)(ws + 2 * szA + 4 * szX);
  float* dinv = (float*)(ws + 2 * szA + 5 * szX);
  float* score= dinv + NN;
  float* vals = score + NN;
  int*   perm = (int*)(vals + NN);

  auto gemm = [&](const float* A, const float* B, float* C, int M, int N,
                  int K, int lda, int ldb, int ldc) {
    dim3 grid((N + BN - 1) / BN, (M + BM - 1) / BM);
    k_gemm<<<grid, 256, 0, stream>>>(A, B, C, M, N, K, lda, ldb, ldc);
  };

  auto gcn = [&](const float* Acur, int n, const float* Xin, int ldxin,
                 int cin, const float* W, int cout, const float* b,
                 float* Xout, int ldxout, bool relu) {
    gemm(Xin, W, XW, n, cout, cin, ldxin, cout, cout);          // XW = X @ W
    k_dinv<<<n, 256, 0, stream>>>(Acur, n, LDA, dinv);
    k_rowscale<<<dim3((cout + 255) / 256, n), 256, 0, stream>>>(XW, dinv, Y,
                                                                cout);
    gemm(Acur, Y, Z, n, cout, n, LDA, cout, cout);              // Z = A @ Y
    k_gcnout<<<dim3((cout + 255) / 256, n), 256, 0, stream>>>(
        Z, Y, dinv, b, Xout, cout, ldxout, relu ? 1 : 0);
  };

  auto augment = [&](const float* Asrc, float* Adst, int n) {
    gemm(Asrc, Asrc, Adst, n, n, n, LDA, LDA, LDA);             // A @ A
    k_augfix<<<dim3((n + 255) / 256, n), 256, 0, stream>>>(Adst, Asrc, n, LDA);
  };

  // 1) dense adjacency
  hipMemsetAsync(bufA, 0, szA, stream);
  k_scatter<<<(EE + 255) / 256, 256, 0, stream>>>(ei, EE, bufA, LDA);

  // 2) augment -> bufB;  GCN0 (relu)
  augment(bufA, bufB, NN);
  gcn(bufB, NN, x_in, CIN, CIN, W0, CH, b0, X1, CH, true);

  // 3) pooling stage 1 (ratio 0.8)
  augment(bufB, bufA, NN);
  k_score<<<(NN + 255) / 256, 256, 0, stream>>>(X1, NN, CH, p1, CH, score);
  k_select<<<(NN + 255) / 256, 256, 0, stream>>>(score, NN, K1, perm, vals);
  k_gatherA<<<dim3((K1 + 255) / 256, K1), 256, 0, stream>>>(bufA, perm, bufB,
                                                            K1, LDA);
  k_gatherx<<<dim3((CH + 255) / 256, K1), 256, 0, stream>>>(X1, CH, perm, vals,
                                                            X2, CH, CH);
  gcn(bufB, K1, X2, CH, CH, W1, CH, b1, X1, CH, true);

  // 4) pooling stage 2 (ratio 0.6)
  augment(bufB, bufA, K1);
  k_score<<<(K1 + 255) / 256, 256, 0, stream>>>(X1, K1, CH, p2, CH, score);
  k_select<<<(K1 + 255) / 256, 256, 0, stream>>>(score, K1, K2, perm, vals);
  k_gatherA<<<dim3((K2 + 255) / 256, K2), 256, 0, stream>>>(bufA, perm, bufB,
                                                            K2, LDA);
  k_gatherx<<<dim3((CH + 255) / 256, K2), 256, 0, stream>>>(X1, CH, perm, vals,
                                                            X2, CH, CH);
  gcn(bufB, K2, X2, CH, CH, W2, CH, b2, X1, CH, true);

  // 5) output GCN (no relu) straight into d_out; batch = zeros after it
  gcn(bufB, K2, X1, CH, CH, Wout, COUT, bout, out, COUT, false);
  hipMemsetAsync(out + (size_t)K2 * COUT, 0, (size_t)K2 * sizeof(float),
                 stream);
}